// GraphClassificationLoss_14757507629507
// MI455X (gfx1250) — compile-verified
//
#include <hip/hip_runtime.h>

#define NUM_GRAPHS 4096
#define WSEG 1024            // per-block LDS segment window (16 KB of LDS)
#define BLOCK 256            // 8 wave32s per block
#define GRID 1024

typedef float v2f __attribute__((ext_vector_type(2)));
typedef float v8f __attribute__((ext_vector_type(8)));

// ---------------------------------------------------------------------------
// Kernel 0: zero the 4096 x 4 float accumulator workspace (SoA layout).
// ---------------------------------------------------------------------------
__global__ void zero_ws_kernel(float* __restrict__ ws, int n) {
    int i = blockIdx.x * blockDim.x + threadIdx.x;
    if (i < n) ws[i] = 0.0f;
}

// ---------------------------------------------------------------------------
// Kernel 1: streaming segmented accumulation.
// ws layout (SoA): [0..4095] sum(e), [4096..] sum(e-shared_e),
//                  [8192..] sum(shared_e*(p0-y)^2), [12288..] sum(bg_e*(p1-(1-y))^2)
// Each block owns a contiguous slab; each wave a contiguous sub-slab; each lane
// 4 contiguous elements per iteration (float4 loads, fully coalesced).
// batch[] is sorted, so lanes keep register accumulators keyed by the current
// segment and flush (rarely) into an LDS window via ds_add_f32; block merges
// the ~few touched segments into global ws with global_atomic_add_f32.
// ---------------------------------------------------------------------------
__global__ void __launch_bounds__(BLOCK) seg_accum_kernel(
    const float* __restrict__ preds,   // (N,2) interleaved
    const float* __restrict__ e,
    const float* __restrict__ se,
    const float* __restrict__ y,
    const int*   __restrict__ batch,
    float*       __restrict__ ws,
    int n, int itersPerWave, int slabBlock)
{
    __shared__ float lE[WSEG];
    __shared__ float lB[WSEG];
    __shared__ float lF[WSEG];
    __shared__ float lG[WSEG];

    float* __restrict__ wsE = ws;
    float* __restrict__ wsB = ws + NUM_GRAPHS;
    float* __restrict__ wsF = ws + 2 * NUM_GRAPHS;
    float* __restrict__ wsG = ws + 3 * NUM_GRAPHS;

    const int tid = threadIdx.x;
    const int blockStart = blockIdx.x * slabBlock;
    if (blockStart >= n) return;                       // uniform per block
    int blockEnd = blockStart + slabBlock;
    if (blockEnd > n) blockEnd = n;

    // Segment range covered by this block's slab (uniform -> scalar loads).
    const int bLo = batch[blockStart];
    const int bHi = batch[blockEnd - 1];

    for (int i = tid; i < WSEG; i += BLOCK) {
        lE[i] = 0.0f; lB[i] = 0.0f; lF[i] = 0.0f; lG[i] = 0.0f;
    }
    __syncthreads();

    const int wave = tid >> 5;
    const int lane = tid & 31;
    const int waveBase = blockStart + wave * (itersPerWave * 128);

    int   curSeg = -1;
    float aE = 0.0f, aB = 0.0f, aF = 0.0f, aG = 0.0f;

    for (int j = 0; j < itersPerWave; ++j) {
        const int idx = waveBase + j * 128 + lane * 4;
        if (idx >= n) break;                           // never taken for exact N

        const float4 e4 = *(const float4*)(e  + idx);
        const float4 s4 = *(const float4*)(se + idx);
        const float4 y4 = *(const float4*)(y  + idx);
        const int4   b4 = *(const int4*)(batch + idx);
        const float4 p04 = *(const float4*)(preds + 2 * idx);       // elems idx, idx+1
        const float4 p14 = *(const float4*)(preds + 2 * idx + 4);   // elems idx+2, idx+3

        const float ev[4] = { e4.x, e4.y, e4.z, e4.w };
        const float sv[4] = { s4.x, s4.y, s4.z, s4.w };
        const float yv[4] = { y4.x, y4.y, y4.z, y4.w };
        const int   bv[4] = { b4.x, b4.y, b4.z, b4.w };
        const float pa[4] = { p04.x, p04.z, p14.x, p14.z };  // preds[:,0]
        const float pb[4] = { p04.y, p04.w, p14.y, p14.w };  // preds[:,1]

        #pragma unroll
        for (int k = 0; k < 4; ++k) {
            const float bge = ev[k] - sv[k];
            const float d0  = pa[k] - yv[k];
            const float d1  = pb[k] - (1.0f - yv[k]);
            const float vF  = sv[k] * d0 * d0;
            const float vG  = bge * d1 * d1;
            const int   seg = bv[k];
            if (seg == curSeg) {
                aE += ev[k]; aB += bge; aF += vF; aG += vG;
            } else {
                if (curSeg >= 0) {                      // flush previous run
                    const unsigned off = (unsigned)(curSeg - bLo);
                    if (off < WSEG) {
                        atomicAdd(&lE[off], aE); atomicAdd(&lB[off], aB);
                        atomicAdd(&lF[off], aF); atomicAdd(&lG[off], aG);
                    } else {                            // pathological escape
                        atomicAdd(&wsE[curSeg], aE); atomicAdd(&wsB[curSeg], aB);
                        atomicAdd(&wsF[curSeg], aF); atomicAdd(&wsG[curSeg], aG);
                    }
                }
                curSeg = seg;
                aE = ev[k]; aB = bge; aF = vF; aG = vG;
            }
        }
    }
    // Final per-lane flush.
    if (curSeg >= 0) {
        const unsigned off = (unsigned)(curSeg - bLo);
        if (off < WSEG) {
            atomicAdd(&lE[off], aE); atomicAdd(&lB[off], aB);
            atomicAdd(&lF[off], aF); atomicAdd(&lG[off], aG);
        } else {
            atomicAdd(&wsE[curSeg], aE); atomicAdd(&wsB[curSeg], aB);
            atomicAdd(&wsF[curSeg], aF); atomicAdd(&wsG[curSeg], aG);
        }
    }
    __syncthreads();

    // Merge the (tiny) touched window into global accumulators.
    int span = bHi - bLo + 1;
    if (span > WSEG) span = WSEG;
    for (int i = tid; i < span; i += BLOCK) {
        atomicAdd(&wsE[bLo + i], lE[i]);
        atomicAdd(&wsB[bLo + i], lB[i]);
        atomicAdd(&wsF[bLo + i], lF[i]);
        atomicAdd(&wsG[bLo + i], lG[i]);
    }
}

// ---------------------------------------------------------------------------
// Kernel 2: finalize with a single wave32. Per-lane partial over 128 graphs,
// then exact f32 cross-lane reduction via V_WMMA_F32_16X16X4_F32 (B = ones):
//   D[m][n] = s_m + s_{m+16};  sum of the 8 D VGPRs per lane gives half-sums,
//   one shfl_xor(16) completes the 32-lane sum.
// ---------------------------------------------------------------------------
__global__ void __launch_bounds__(32) finalize_kernel(
    const float* __restrict__ ws, float* __restrict__ out)
{
    const float* __restrict__ wsE = ws;
    const float* __restrict__ wsB = ws + NUM_GRAPHS;
    const float* __restrict__ wsF = ws + 2 * NUM_GRAPHS;
    const float* __restrict__ wsG = ws + 3 * NUM_GRAPHS;

    const int lane = threadIdx.x;
    float s = 0.0f;
    for (int g = lane; g < NUM_GRAPHS; g += 32) {
        const float fg = wsF[g];
        const float bg = wsG[g];
        const float fs = 1.0f + wsE[g];
        const float bs = 1.0f + wsB[g];
        s += fg / fs + bg / bs;
    }

    // A (16x4 f32): VGPR0 = lane partial (K=0 for lanes 0-15, K=2 for 16-31),
    // VGPR1 = 0. B (4x16 f32): all ones. C = 0.
    v2f a; a.x = s;    a.y = 0.0f;
    v2f b; b.x = 1.0f; b.y = 1.0f;
    v8f c = {};
    c = __builtin_amdgcn_wmma_f32_16x16x4_f32(
            /*neg_a=*/false, a, /*neg_b=*/false, b,
            /*c_mod=*/(short)0, c, /*reuse_a=*/false, /*reuse_b=*/false);

    float t = c[0] + c[1] + c[2] + c[3] + c[4] + c[5] + c[6] + c[7];
    t += __shfl_xor(t, 16, 32);
    if (lane == 0) out[0] = t;
}

// ---------------------------------------------------------------------------
extern "C" void kernel_launch(void* const* d_in, const int* in_sizes, int n_in,
                              void* d_out, int out_size, void* d_ws, size_t ws_size,
                              hipStream_t stream) {
    (void)n_in; (void)out_size; (void)ws_size;
    const float* preds = (const float*)d_in[0];
    const float* e     = (const float*)d_in[1];
    const float* se    = (const float*)d_in[2];
    const float* y     = (const float*)d_in[3];
    const int*   batch = (const int*)d_in[4];
    const int n = in_sizes[1];          // N elements (preds is 2N)

    float* ws  = (float*)d_ws;          // 4 * NUM_GRAPHS floats (64 KB)
    float* out = (float*)d_out;

    const int wsElems = 4 * NUM_GRAPHS;
    zero_ws_kernel<<<(wsElems + 255) / 256, 256, 0, stream>>>(ws, wsElems);

    const int grid = GRID;
    const long long totalWaves = (long long)grid * (BLOCK / 32);
    long long perWave = (n + totalWaves - 1) / totalWaves;
    perWave = ((perWave + 127) / 128) * 128;            // multiple of 128 elems
    const int itersPerWave = (int)(perWave / 128);
    const int slabBlock    = (int)(perWave * (BLOCK / 32));

    seg_accum_kernel<<<grid, BLOCK, 0, stream>>>(
        preds, e, se, y, batch, ws, n, itersPerWave, slabBlock);

    finalize_kernel<<<1, 32, 0, stream>>>(ws, out);
}